// PositionMultiHeadAttention_69295002354136
// MI455X (gfx1250) — compile-verified
//
#include <hip/hip_runtime.h>
#include <math.h>

typedef __attribute__((ext_vector_type(2))) float v2f;
typedef __attribute__((ext_vector_type(4))) float v4f;
typedef __attribute__((ext_vector_type(8))) float v8f;
typedef __attribute__((ext_vector_type(4))) int   v4i;

#define N_SEQ  1024
#define BATCH  8
#define EMB    1024
#define HEADS  16
#define HDIM   64
#define STACKS 8
#define ZDIM   (BATCH * HEADS) /* 128 */

#if defined(__has_builtin)
#if __has_builtin(__builtin_amdgcn_global_load_async_to_lds_b128)
#define HAVE_ASYNC_LDS 1
#endif
#endif
#ifndef HAVE_ASYNC_LDS
#define HAVE_ASYNC_LDS 0
#endif

static __device__ __forceinline__ v8f wmma_f32(v2f a, v2f b, v8f c) {
  // D = A(16x4,f32) * B(4x16,f32) + C(16x16,f32); exact fp32 matrix op on CDNA5
  return __builtin_amdgcn_wmma_f32_16x16x4_f32(false, a, false, b, (short)0, c,
                                               false, false);
}

static __device__ __forceinline__ void wait_async_lds() {
#if HAVE_ASYNC_LDS
#if __has_builtin(__builtin_amdgcn_s_wait_asynccnt)
  __builtin_amdgcn_s_wait_asynccnt(0);
#else
  asm volatile("s_wait_asynccnt 0x0" ::: "memory");
#endif
#endif
}

// ---------------------------------------------------------------------------
// Kernel 1/5: fp32 GEMM with bias, out[m][c] = sum_k A[m][k]*W[c][k] + bias[c]
// M=8192, N=1024, K=1024.  128 threads / block (4 waves); block tile is
// 16(M) x 64(N); K consumed in 64-wide chunks double-buffered through LDS via
// GLOBAL_LOAD_ASYNC_TO_LDS_B128 (ASYNCcnt) when available.  Each wave owns one
// 16x16 N-tile and runs V_WMMA_F32_16X16X4_F32 over the staged chunk.
// vlayout==1 scatters the result into v[z][n][hd] for the attention kernel.
// ---------------------------------------------------------------------------
__global__ void gemm16_kernel(const float* __restrict__ A,
                              const float* __restrict__ W,
                              const float* __restrict__ bias,
                              float* __restrict__ out, int vlayout) {
  // Row stride 68 floats: lane rl hits bank 4*rl (+k) -> conflict-free halves,
  // rows stay 16B aligned for b128 stores.
  __shared__ float At[2][16][68];
  __shared__ float Wt[2][64][68];

  const int tid  = threadIdx.x;          // 128 threads
  const int wave = tid >> 5;
  const int lane = tid & 31;
  const int half = lane >> 4;            // K sub-phase (per ISA A/B layout)
  const int rl   = lane & 15;
  const int mtile = blockIdx.x;          // 512 tiles of 16 rows
  const int ng    = blockIdx.y;          // 16 groups of 64 cols
  const int m0 = mtile * 16;
  const int c0 = ng * 64;
  const int c  = c0 + wave * 16 + rl;

#if HAVE_ASYNC_LDS
#define ASYNC_CP16(GSRC, LDST)                                                 \
  __builtin_amdgcn_global_load_async_to_lds_b128(                              \
      (__attribute__((address_space(1))) v4i*)(uintptr_t)(GSRC),               \
      (__attribute__((address_space(3))) v4i*)(unsigned)(uintptr_t)(LDST), 0,  \
      0)
#else
#define ASYNC_CP16(GSRC, LDST) ((void)0)
#endif

#define STAGE_CHUNK(BUF, K0)                                                   \
  do {                                                                         \
    /* A chunk: 16 rows x 64 floats = 256 float4, 2 per thread */              \
    _Pragma("unroll") for (int r = 0; r < 2; ++r) {                            \
      const int q = tid + 128 * r;                                             \
      const int row = q >> 4, col = (q & 15) * 4;                              \
      const float* gsrc = A + (size_t)(m0 + row) * EMB + (K0) + col;           \
      float* ldst = &At[BUF][row][col];                                        \
      if (HAVE_ASYNC_LDS) {                                                    \
        ASYNC_CP16(gsrc, ldst);                                                \
      } else {                                                                 \
        *(v4f*)ldst = *(const v4f*)gsrc;                                       \
      }                                                                        \
    }                                                                          \
    /* W chunk: 64 rows x 64 floats = 1024 float4, 8 per thread */             \
    _Pragma("unroll") for (int r = 0; r < 8; ++r) {                            \
      const int q = tid + 128 * r;                                             \
      const int row = q >> 4, col = (q & 15) * 4;                              \
      const float* gsrc = W + (size_t)(c0 + row) * EMB + (K0) + col;           \
      float* ldst = &Wt[BUF][row][col];                                        \
      if (HAVE_ASYNC_LDS) {                                                    \
        ASYNC_CP16(gsrc, ldst);                                                \
      } else {                                                                 \
        *(v4f*)ldst = *(const v4f*)gsrc;                                       \
      }                                                                        \
    }                                                                          \
  } while (0)

  v8f acc = {0.f, 0.f, 0.f, 0.f, 0.f, 0.f, 0.f, 0.f};

  STAGE_CHUNK(0, 0);
  int buf = 0;
#pragma unroll 1
  for (int kc = 0; kc < EMB / 64; ++kc) {
    wait_async_lds();
    __syncthreads();                       // chunk `buf` fully in LDS
    if (kc + 1 < EMB / 64) STAGE_CHUNK(buf ^ 1, (kc + 1) * 64);

#pragma unroll
    for (int kk = 0; kk < 64; kk += 4) {
      const int ka = kk + 2 * half;
      v2f Af, Bf;
      Af.x = At[buf][rl][ka];
      Af.y = At[buf][rl][ka + 1];
      Bf.x = Wt[buf][wave * 16 + rl][ka];
      Bf.y = Wt[buf][wave * 16 + rl][ka + 1];
      acc = wmma_f32(Af, Bf, acc);
    }
    __syncthreads();                       // readers done before next overwrite
    buf ^= 1;
  }

  const float bc = bias[c];
#pragma unroll
  for (int i8 = 0; i8 < 8; ++i8) {
    const int mm = m0 + i8 + 8 * half;     // D row (ISA C/D layout)
    const float val = acc[i8] + bc;
    if (vlayout) {
      const int n = mm >> 3, bb = mm & 7;  // query rows are (n, b)
      const int hh = c >> 6, d = c & 63;   // e = h*64 + d
      out[(((size_t)(bb * HEADS + hh)) * N_SEQ + n) * HDIM + d] = val;
    } else {
      out[(size_t)mm * EMB + c] = val;
    }
  }
#undef STAGE_CHUNK
#undef ASYNC_CP16
}

// ---------------------------------------------------------------------------
// Kernel 2/5: per-(z,row) softmax statistics (max, 1/sum_exp).
// One wave per (b,i) row; all 16 heads computed from one adj sweep (online).
// ---------------------------------------------------------------------------
__global__ void rowstats_kernel(const float* __restrict__ adj,
                                const float* __restrict__ pos_w,
                                const float* __restrict__ pos_b,
                                float* __restrict__ mx_ws,
                                float* __restrict__ is_ws) {
  __shared__ float pw[HEADS * STACKS];
  __shared__ float pb[HEADS];
  const int tid = threadIdx.x;           // 256 threads = 8 waves
  if (tid < HEADS * STACKS) pw[tid] = pos_w[tid];
  if (tid < HEADS) pb[tid] = pos_b[tid];
  __syncthreads();

  const int gwave = (int)((blockIdx.x * blockDim.x + tid) >> 5);
  const int lane  = tid & 31;
  if (gwave >= BATCH * N_SEQ) return;
  const int b = gwave >> 10;
  const int i = gwave & 1023;

  float m[HEADS], s[HEADS];
#pragma unroll
  for (int h = 0; h < HEADS; ++h) { m[h] = -3.0e38f; s[h] = 0.f; }

  const float* base =
      adj + (size_t)(b * STACKS) * N_SEQ * N_SEQ + (size_t)i * N_SEQ;
  for (int j = lane; j < N_SEQ; j += 32) {
    float a[STACKS];
#pragma unroll
    for (int st = 0; st < STACKS; ++st)
      a[st] = base[(size_t)st * N_SEQ * N_SEQ + j];
#pragma unroll
    for (int h = 0; h < HEADS; ++h) {
      float sc = pb[h];
#pragma unroll
      for (int st = 0; st < STACKS; ++st)
        sc = fmaf(a[st], pw[h * STACKS + st], sc);
      const float mo = m[h];
      const float mn = fmaxf(mo, sc);
      s[h] = s[h] * __expf(mo - mn) + __expf(sc - mn);
      m[h] = mn;
    }
  }

  // wave32 butterfly combine of (max, sum) pairs
#pragma unroll
  for (int off = 16; off >= 1; off >>= 1) {
#pragma unroll
    for (int h = 0; h < HEADS; ++h) {
      const float mo = __shfl_xor(m[h], off, 32);
      const float so = __shfl_xor(s[h], off, 32);
      const float mn = fmaxf(m[h], mo);
      s[h] = s[h] * __expf(m[h] - mn) + so * __expf(mo - mn);
      m[h] = mn;
    }
  }

  if (lane == 0) {
#pragma unroll
    for (int h = 0; h < HEADS; ++h) {
      mx_ws[(size_t)(b * HEADS + h) * N_SEQ + i] = m[h];
      is_ws[(size_t)(b * HEADS + h) * N_SEQ + i] = 1.0f / s[h];
    }
  }
}

// ---------------------------------------------------------------------------
// Kernel 3/5: head-averaged attention weights, written straight to output.
// avg[b][i][j] = (1/16) * sum_h exp(score_h - mx) * invsum.  One block per
// (b,i) row; pure elementwise, adj read once.
// ---------------------------------------------------------------------------
__global__ void avg_weights_kernel(const float* __restrict__ adj,
                                   const float* __restrict__ pos_w,
                                   const float* __restrict__ pos_b,
                                   const float* __restrict__ mx_ws,
                                   const float* __restrict__ is_ws,
                                   float* __restrict__ out_avg) {
  __shared__ float pw[HEADS * STACKS];
  __shared__ float pb[HEADS];
  __shared__ float mx[HEADS];
  __shared__ float is[HEADS];
  const int tid = threadIdx.x;           // 256
  const int b = blockIdx.x >> 10;
  const int i = blockIdx.x & 1023;
  if (tid < HEADS * STACKS) pw[tid] = pos_w[tid];
  if (tid < HEADS) {
    pb[tid] = pos_b[tid];
    mx[tid] = mx_ws[(size_t)(b * HEADS + tid) * N_SEQ + i];
    is[tid] = is_ws[(size_t)(b * HEADS + tid) * N_SEQ + i];
  }
  __syncthreads();

  const float* base =
      adj + (size_t)(b * STACKS) * N_SEQ * N_SEQ + (size_t)i * N_SEQ;
  for (int j = tid; j < N_SEQ; j += 256) {
    float a[STACKS];
#pragma unroll
    for (int st = 0; st < STACKS; ++st)
      a[st] = base[(size_t)st * N_SEQ * N_SEQ + j];
    float acc = 0.f;
#pragma unroll
    for (int h = 0; h < HEADS; ++h) {
      float sc = pb[h];
#pragma unroll
      for (int st = 0; st < STACKS; ++st)
        sc = fmaf(a[st], pw[h * STACKS + st], sc);
      acc += __expf(sc - mx[h]) * is[h];
    }
    out_avg[((size_t)b * N_SEQ + i) * N_SEQ + j] = acc * (1.0f / HEADS);
  }
}

// ---------------------------------------------------------------------------
// Kernel 4/5: P = softmax(scores_z) row-tile staged in LDS (skewed to avoid
// stride-1024 bank conflicts), then WMMA P(16x1024) * V(1024x64).  One block
// (4 waves) per (z, 16-row tile); each wave owns one 16-col N-tile of hd.
// Result scattered into [n][b][e] for the final out-proj GEMM.
// ---------------------------------------------------------------------------
__global__ void attn_v_kernel(const float* __restrict__ adj,
                              const float* __restrict__ pos_w,
                              const float* __restrict__ pos_b,
                              const float* __restrict__ mx_ws,
                              const float* __restrict__ is_ws,
                              const float* __restrict__ v_ws,
                              float* __restrict__ ao_ws) {
  __shared__ float P[16 * 1024];         // 64 KB, skewed addressing
  const int tid = threadIdx.x;           // 128
  const int z  = blockIdx.x;             // 0..127  (z = b*16 + h)
  const int rt = blockIdx.y;             // 0..63
  const int b = z >> 4, h = z & 15;
  const int r0 = rt * 16;

  float pw8[STACKS];
#pragma unroll
  for (int st = 0; st < STACKS; ++st) pw8[st] = pos_w[h * STACKS + st];
  const float pbh = pos_b[h];

  const float* base = adj + (size_t)(b * STACKS) * N_SEQ * N_SEQ;
  for (int idx = tid; idx < 16 * N_SEQ; idx += 128) {
    const int r = idx >> 10, j = idx & 1023;
    const int row = r0 + r;
    float sc = pbh;
#pragma unroll
    for (int st = 0; st < STACKS; ++st)
      sc = fmaf(base[((size_t)st * N_SEQ + row) * N_SEQ + j], pw8[st], sc);
    const float p = __expf(sc - mx_ws[(size_t)z * N_SEQ + row]) *
                    is_ws[(size_t)z * N_SEQ + row];
    P[(r << 10) + ((j + r) & 1023)] = p;   // skew by row
  }
  __syncthreads();

  const int wave = tid >> 5, lane = tid & 31;
  const int half = lane >> 4, rl = lane & 15;
  const int col = wave * 16 + rl;          // hd column 0..63
  const float* vz = v_ws + (size_t)z * N_SEQ * HDIM;

  v8f acc = {0.f, 0.f, 0.f, 0.f, 0.f, 0.f, 0.f, 0.f};
  for (int k0 = 0; k0 < N_SEQ; k0 += 4) {
    const int ka = k0 + 2 * half;
    v2f A, B;
    A.x = P[(rl << 10) + ((ka + rl) & 1023)];
    A.y = P[(rl << 10) + ((ka + 1 + rl) & 1023)];
    B.x = vz[(size_t)ka * HDIM + col];
    B.y = vz[(size_t)(ka + 1) * HDIM + col];
    acc = wmma_f32(A, B, acc);
  }

#pragma unroll
  for (int i8 = 0; i8 < 8; ++i8) {
    const int n = r0 + i8 + 8 * half;
    ao_ws[((size_t)n * BATCH + b) * EMB + h * HDIM + col] = acc[i8];
  }
}

// ---------------------------------------------------------------------------
extern "C" void kernel_launch(void* const* d_in, const int* in_sizes, int n_in,
                              void* d_out, int out_size, void* d_ws,
                              size_t ws_size, hipStream_t stream) {
  (void)in_sizes; (void)n_in; (void)out_size; (void)ws_size;
  const float* query = (const float*)d_in[0];  // [1024, 8, 1024]
  const float* adj   = (const float*)d_in[1];  // [8, 8, 1024, 1024]
  const float* pos_w = (const float*)d_in[2];  // [16, 8]
  const float* pos_b = (const float*)d_in[3];  // [16]
  const float* ipw   = (const float*)d_in[4];  // [1024, 1024]
  const float* ipb   = (const float*)d_in[5];  // [1024]
  const float* outw  = (const float*)d_in[6];  // [1024, 1024]
  const float* outb  = (const float*)d_in[7];  // [1024]

  float* out      = (float*)d_out;
  float* attn_out = out;                                   // [1024,8,1024]
  float* avg_out  = out + (size_t)N_SEQ * BATCH * EMB;     // [8,1024,1024]

  float* ws    = (float*)d_ws;
  float* v_ws  = ws;                                       // 8M  floats
  float* mx_ws = v_ws + (size_t)ZDIM * N_SEQ * HDIM;       // 128K floats
  float* is_ws = mx_ws + (size_t)ZDIM * N_SEQ;             // 128K floats
  float* ao_ws = is_ws + (size_t)ZDIM * N_SEQ;             // 8M  floats

  // V projection: v[z][n][hd] = (query @ ipw^T + ipb) reshaped
  gemm16_kernel<<<dim3(512, 16), 128, 0, stream>>>(query, ipw, ipb, v_ws, 1);
  // Softmax row statistics for all 128 (b,h) score matrices
  rowstats_kernel<<<dim3(1024), 256, 0, stream>>>(adj, pos_w, pos_b, mx_ws,
                                                  is_ws);
  // Head-averaged attention weights (second output)
  avg_weights_kernel<<<dim3(8192), 256, 0, stream>>>(adj, pos_w, pos_b, mx_ws,
                                                     is_ws, avg_out);
  // attn @ v  (per-z WMMA, P staged in LDS)
  attn_v_kernel<<<dim3(128, 64), 128, 0, stream>>>(adj, pos_w, pos_b, mx_ws,
                                                   is_ws, v_ws, ao_ws);
  // Output projection (first output)
  gemm16_kernel<<<dim3(512, 16), 128, 0, stream>>>(ao_ws, outw, outb, attn_out,
                                                   0);
}